// SSIM_644245094434
// MI455X (gfx1250) — compile-verified
//
#include <hip/hip_runtime.h>

// ---------------------------------------------------------------------------
// Fused separable-Gaussian SSIM loss for MI455X (gfx1250, wave32).
// BOTH separable passes run on the matrix pipe as banded-Toeplitz f32 WMMA
// (V_WMMA_F32_16X16X4_F32, 11-tap band padded K=26->28, 7 chunks):
//   h-pass:  D(16x16) = Xpatch(16x28) x Wband(28x16)   (A = image, B = band)
//   v-pass:  D(16x16) = Wband(16x28)  x Hcols(28x16)   (A = band, B = h data)
// The per-lane band operand w[j - (lane&15)] is identical for both roles.
// SSIM map + deterministic reduction fused; kernel 2 folds 12288 partials.
// ---------------------------------------------------------------------------

#define KS   11
#define HH   512
#define WW   512
#define HO   502
#define WO   502
#define BR   32          // output rows per workgroup
#define BC   64          // output cols per workgroup
#define IR   44          // h-plane rows: BR + (28 - 16) K-padding
#define IC   76          // staged cols: BC + 10 taps + 2 K-pad
#define SXP  78          // staged row stride (even -> every row 8B aligned)
#define SHP  84          // h-plane row stride (conflict-free for both passes)
#define NP   5
#define GX   8
#define GY   16
#define GZ   96          // 32 batch * 3 channels
#define NWG  (GX*GY*GZ)  // 12288 partials
#define NELEM_INV (1.0f / 24192384.0f)  // 1 / (32*3*502*502)

typedef float v2f __attribute__((ext_vector_type(2)));
typedef float v8f __attribute__((ext_vector_type(8)));

__global__ __launch_bounds__(256)
void ssim_fused_kernel(const float* __restrict__ X, const float* __restrict__ Y,
                       const float* __restrict__ win, float* __restrict__ partial)
{
    __shared__ float sW[16];                            // 1D gaussian (row sums)
    __shared__ __align__(16) float sX[IR * SXP];
    __shared__ __align__(16) float sY[IR * SXP];
    __shared__ __align__(16) float sH[NP * IR * SHP];   // h-filtered x,y,xx,yy,xy
    __shared__ float sR[256];

    const int tid = threadIdx.x;
    const int bx = blockIdx.x, by = blockIdx.y, bz = blockIdx.z;
    const int C0 = bx * BC;
    const int R0 = by * BR;
    const size_t pbase = (size_t)bz * (size_t)(HH * WW);

    // 1D kernel from 2D win (win is rank-1, normalized: row sums == w1d)
    if (tid < KS) {
        float s = 0.f;
        #pragma unroll
        for (int j = 0; j < KS; ++j) s += win[tid * KS + j];
        sW[tid] = s;
    }

    // Stage clamped X/Y input tiles (44 x 76)
    const float* Xp = X + pbase;
    const float* Yp = Y + pbase;
    for (int idx = tid; idx < IR * IC; idx += 256) {
        int r = idx / IC;
        int c = idx - r * IC;
        int gr = min(R0 + r, HH - 1);
        int gc = min(C0 + c, WW - 1);
        size_t g = (size_t)gr * WW + gc;
        sX[r * SXP + c] = Xp[g];
        sY[r * SXP + c] = Yp[g];
    }
    __syncthreads();

    const int lane = tid & 31;
    const int wv   = tid >> 5;             // wave 0..7
    const int m    = lane & 15;
    const int hi   = lane >> 4;            // lane half
    const int koff = hi * 2;               // K sub-offset per lane half

    // Banded gaussian operand: entry = w[(4kk+koff+p) - m], 0 if outside band.
    // Serves as B (h-pass, m == out col) and as A (v-pass, m == out row).
    v2f gw[7];
    #pragma unroll
    for (int kk = 0; kk < 7; ++kk) {
        int j0 = kk * 4 + koff;
        int i0 = j0 - m, i1 = i0 + 1;
        int c0i = min(max(i0, 0), KS - 1);
        int c1i = min(max(i1, 0), KS - 1);
        float v0 = sW[c0i], v1 = sW[c1i];
        gw[kk][0] = (i0 == c0i) ? v0 : 0.f;
        gw[kk][1] = (i1 == c1i) ? v1 : 0.f;
    }

    // -------- Horizontal pass via WMMA: 12 tiles (rows {0,16,28} x 4 cols) ---
    for (int t = wv; t < 12; t += 8) {
        const int rsel = t >> 2;                     // 0,1,2
        const int r0   = (rsel == 0) ? 0 : ((rsel == 1) ? 16 : 28);
        const int cb   = (t & 3) * 16;
        const float* rowX = sX + (r0 + m) * SXP + cb + koff;
        const float* rowY = sY + (r0 + m) * SXP + cb + koff;

        v8f d0 = {}, d1 = {}, d2 = {}, d3 = {}, d4 = {};
        #pragma unroll
        for (int kk = 0; kk < 7; ++kk) {
            v2f ax = *(const v2f*)(rowX + kk * 4);   // ds_load_b64 (8B aligned)
            v2f ay = *(const v2f*)(rowY + kk * 4);
            v2f axx, ayy, axy;
            axx[0] = ax[0] * ax[0]; axx[1] = ax[1] * ax[1];
            ayy[0] = ay[0] * ay[0]; ayy[1] = ay[1] * ay[1];
            axy[0] = ax[0] * ay[0]; axy[1] = ax[1] * ay[1];
            d0 = __builtin_amdgcn_wmma_f32_16x16x4_f32(false, ax,  false, gw[kk], (short)0, d0, false, false);
            d1 = __builtin_amdgcn_wmma_f32_16x16x4_f32(false, ay,  false, gw[kk], (short)0, d1, false, false);
            d2 = __builtin_amdgcn_wmma_f32_16x16x4_f32(false, axx, false, gw[kk], (short)0, d2, false, false);
            d3 = __builtin_amdgcn_wmma_f32_16x16x4_f32(false, ayy, false, gw[kk], (short)0, d3, false, false);
            d4 = __builtin_amdgcn_wmma_f32_16x16x4_f32(false, axy, false, gw[kk], (short)0, d4, false, false);
        }

        // Scatter D into sH. D layout: lane half hi -> rows i + 8*hi, col m.
        // Tile r0==28 overlaps rows 28..31 already written by tile r0==16.
        float* hdst = sH + (r0 + 8 * hi) * SHP + cb + m;
        #pragma unroll
        for (int i = 0; i < 8; ++i) {
            if (r0 != 28 || (8 * hi + i) >= 4) {
                hdst[i * SHP + 0 * IR * SHP] = d0[i];
                hdst[i * SHP + 1 * IR * SHP] = d1[i];
                hdst[i * SHP + 2 * IR * SHP] = d2[i];
                hdst[i * SHP + 3 * IR * SHP] = d3[i];
                hdst[i * SHP + 4 * IR * SHP] = d4[i];
            }
        }
    }
    __syncthreads();

    // -------- Vertical pass via WMMA: D = Wband(16x28) x Hcols(28x16) --------
    const int trow = (wv >> 2) * 16;       // 0 | 16  (local tile row)
    const int tcol = (wv & 3) * 16;        // 0..48   (local tile col)

    v8f acc0 = {}, acc1 = {}, acc2 = {}, acc3 = {}, acc4 = {};
    #pragma unroll
    for (int kk = 0; kk < 7; ++kk) {
        const int rA = trow + kk * 4 + koff;          // <= 42 (+1 -> 43 < 44)
        const float* hb = sH + rA * SHP + tcol + m;
        v2f b0, b1, b2, b3, b4;
        b0[0] = hb[0 * IR * SHP]; b0[1] = hb[0 * IR * SHP + SHP];
        b1[0] = hb[1 * IR * SHP]; b1[1] = hb[1 * IR * SHP + SHP];
        b2[0] = hb[2 * IR * SHP]; b2[1] = hb[2 * IR * SHP + SHP];
        b3[0] = hb[3 * IR * SHP]; b3[1] = hb[3 * IR * SHP + SHP];
        b4[0] = hb[4 * IR * SHP]; b4[1] = hb[4 * IR * SHP + SHP];
        acc0 = __builtin_amdgcn_wmma_f32_16x16x4_f32(false, gw[kk], false, b0, (short)0, acc0, false, false);
        acc1 = __builtin_amdgcn_wmma_f32_16x16x4_f32(false, gw[kk], false, b1, (short)0, acc1, false, false);
        acc2 = __builtin_amdgcn_wmma_f32_16x16x4_f32(false, gw[kk], false, b2, (short)0, acc2, false, false);
        acc3 = __builtin_amdgcn_wmma_f32_16x16x4_f32(false, gw[kk], false, b3, (short)0, acc3, false, false);
        acc4 = __builtin_amdgcn_wmma_f32_16x16x4_f32(false, gw[kk], false, b4, (short)0, acc4, false, false);
    }

    // -------- SSIM map + masked local sum ------------------------------------
    const float C1c = 6.5025f;    // (0.01*255)^2
    const float C2c = 58.5225f;   // (0.03*255)^2
    float lsum = 0.f;
    const int outc  = C0 + tcol + m;
    const int rbase = R0 + trow + hi * 8;  // D layout: M = i + 8*hi
    #pragma unroll
    for (int i = 0; i < 8; ++i) {
        float mu1 = acc0[i], mu2 = acc1[i];
        float mu1s = mu1 * mu1, mu2s = mu2 * mu2, mu12 = mu1 * mu2;
        float s1  = acc2[i] - mu1s;
        float s2  = acc3[i] - mu2s;
        float s12 = acc4[i] - mu12;
        float cs   = fmaxf((2.f * s12 + C2c) / (s1 + s2 + C2c), 0.f);
        float ssim = ((2.f * mu12 + C1c) / (mu1s + mu2s + C1c)) * cs;
        lsum += ((rbase + i) < HO && outc < WO) ? ssim : 0.f;
    }

    // -------- Deterministic workgroup reduction ------------------------------
    sR[tid] = lsum;
    __syncthreads();
    #pragma unroll
    for (int s = 128; s > 0; s >>= 1) {
        if (tid < s) sR[tid] += sR[tid + s];
        __syncthreads();
    }
    if (tid == 0) partial[(bz * GY + by) * GX + bx] = sR[0];
}

__global__ __launch_bounds__(256)
void ssim_finalize_kernel(const float* __restrict__ partial, float* __restrict__ out)
{
    __shared__ float sR[256];
    float s = 0.f;
    for (int i = threadIdx.x; i < NWG; i += 256) s += partial[i];  // fixed order
    sR[threadIdx.x] = s;
    __syncthreads();
    #pragma unroll
    for (int k = 128; k > 0; k >>= 1) {
        if (threadIdx.x < k) sR[threadIdx.x] += sR[threadIdx.x + k];
        __syncthreads();
    }
    if (threadIdx.x == 0) out[0] = 1.0f - sR[0] * NELEM_INV;
}

extern "C" void kernel_launch(void* const* d_in, const int* in_sizes, int n_in,
                              void* d_out, int out_size, void* d_ws, size_t ws_size,
                              hipStream_t stream)
{
    const float* X   = (const float*)d_in[0];
    const float* Y   = (const float*)d_in[1];
    const float* win = (const float*)d_in[2];   // (3,1,11,11); channel 0 used
    float* partial = (float*)d_ws;              // NWG floats (48 KB)
    float* out = (float*)d_out;

    dim3 grid(GX, GY, GZ);
    hipLaunchKernelGGL(ssim_fused_kernel, grid, dim3(256), 0, stream, X, Y, win, partial);
    hipLaunchKernelGGL(ssim_finalize_kernel, dim3(1), dim3(256), 0, stream, partial, out);
}